// UniGINLayer_17892833755481
// MI455X (gfx1250) — compile-verified
//
#include <hip/hip_runtime.h>

typedef __attribute__((ext_vector_type(2))) float v2f;
typedef __attribute__((ext_vector_type(8))) float v8f;

#define CDIM 256

// ---------------- zero output buffer (poisoned by harness) ----------------
__global__ void zero4_kernel(float4* __restrict__ p, long long n4) {
  long long i = (long long)blockIdx.x * blockDim.x + threadIdx.x;
  long long stride = (long long)gridDim.x * blockDim.x;
  float4 z = make_float4(0.f, 0.f, 0.f, 0.f);
  for (; i < n4; i += stride) p[i] = z;
}

// ---------------- generic row gather + scatter-add -----------------------
// dst[dst_idx[i], :] += src[src_idx[i], :]   for i in [0, nnz)
// One block per nnz entry; one channel per lane (fully coalesced 1KB row).
// f32 atomics resolve in L2 (dst arrays are L2-resident on MI455X: 51/102 MB).
__global__ __launch_bounds__(CDIM)
void scatter_add_rows(const float* __restrict__ src,
                      const int* __restrict__ src_idx,
                      const int* __restrict__ dst_idx,
                      float* __restrict__ dst, int nnz) {
  int i = blockIdx.x;
  if (i >= nnz) return;
  int s = src_idx[i];
  int d = dst_idx[i];
  int c = threadIdx.x;
  float v = src[(long long)s * CDIM + c];
  unsafeAtomicAdd(&dst[(long long)d * CDIM + c], v);  // global_atomic_add_f32
}

// ---------------- GIN update: out = (x0 + m) @ W^T + b, via fp32 WMMA ------
// Block = 512 threads = 16 waves. Block owns 16 node rows (full 256 channels).
// Stage A = x0 + m into LDS once, then wave w computes output cols [16w,16w+16)
// with 64 chained v_wmma_f32_16x16x4_f32 (K = 256). In-place safe: all reads of
// this block's rows happen before its writes; rows disjoint across blocks.
__global__ __launch_bounds__(512)
void gin_gemm_wmma(const float* __restrict__ x0,
                   const float* __restrict__ m,
                   const float* __restrict__ W,      // (C, C) row-major: W[j][k]
                   const float* __restrict__ bias,
                   float* __restrict__ out, int nrows) {
  __shared__ float At[16][CDIM + 4];  // +4 pad: row stride 260 -> bank offset 4/row
  const int row0 = blockIdx.x * 16;
  const int tid  = threadIdx.x;

  // Cooperative load of the 16x256 A tile (x0 + m)
  for (int i = tid; i < 16 * CDIM; i += 512) {
    int r = i >> 8;
    int k = i & (CDIM - 1);
    int row = row0 + r;
    float v = 0.f;
    if (row < nrows) {
      long long off = (long long)row * CDIM + k;
      v = x0[off] + m[off];
    }
    At[r][k] = v;
  }
  __syncthreads();

  const int wave    = tid >> 5;
  const int lane    = tid & 31;
  const int colBase = wave * 16;          // 16 waves x 16 cols = 256 cols
  const int mrow    = lane & 15;          // A: M index / B: N index / D: col
  const int koff    = (lane >> 4) << 1;   // lanes 0-15: K pair {0,1}; 16-31: {2,3}

  const float* wrow = W + (long long)(colBase + mrow) * CDIM;  // B col = W row

  v8f acc = {};
  #pragma unroll 8
  for (int k0 = 0; k0 < CDIM; k0 += 4) {
    int kb = k0 + koff;
    v2f a, b;
    a.x = At[mrow][kb];
    a.y = At[mrow][kb + 1];
    b.x = wrow[kb];          // B[kb][col]   = W[col][kb]
    b.y = wrow[kb + 1];      // B[kb+1][col] = W[col][kb+1]
    // D = A(16x4) x B(4x16) + C, all fp32 (exact precision match to reference)
    acc = __builtin_amdgcn_wmma_f32_16x16x4_f32(
        /*neg_a=*/false, a, /*neg_b=*/false, b,
        /*c_mod=*/(short)0, acc, /*reuse_a=*/false, /*reuse_b=*/false);
  }

  // D layout: VGPR i -> row (i) for lanes 0-15, row (i+8) for lanes 16-31
  const float bv    = bias[colBase + mrow];
  const int   rbase = row0 + ((lane >> 4) << 3);
  #pragma unroll
  for (int i = 0; i < 8; i++) {
    int row = rbase + i;
    if (row < nrows)
      out[(long long)row * CDIM + colBase + mrow] = acc[i] + bv;
  }
}

extern "C" void kernel_launch(void* const* d_in, const int* in_sizes, int n_in,
                              void* d_out, int out_size, void* d_ws, size_t ws_size,
                              hipStream_t stream) {
  const float* x0       = (const float*)d_in[0];
  const int*   node_idx = (const int*)d_in[1];
  const int*   edge_idx = (const int*)d_in[2];
  const float* W        = (const float*)d_in[3];
  const float* bias     = (const float*)d_in[4];
  float*       out      = (float*)d_out;

  const int C   = in_sizes[4];            // 256
  const int N   = in_sizes[0] / C;        // 100000 nodes
  const int NNZ = in_sizes[1];            // 800000 incidences

  float* x0_out = out;                          // (N, C) — also m accumulator
  float* x1     = out + (long long)N * C;       // (E, C) — second output

  // 1) zero both output regions (x1 and m are built by atomics)
  {
    long long n4 = (long long)out_size / 4;     // out_size divisible by 4 (C=256)
    int blocks = 4096;
    zero4_kernel<<<blocks, 256, 0, stream>>>((float4*)out, n4);
  }

  // 2) vertex2edge: x1[edge_idx[i]] += x0[node_idx[i]]
  scatter_add_rows<<<NNZ, CDIM, 0, stream>>>(x0, node_idx, edge_idx, x1, NNZ);

  // 3) edge2vertex: m[node_idx[i]] += x1[edge_idx[i]]   (m lives in x0_out)
  scatter_add_rows<<<NNZ, CDIM, 0, stream>>>(x1, edge_idx, node_idx, x0_out, NNZ);

  // 4) GIN update in place: x0_out = (x0 + m) @ W^T + b
  gin_gemm_wmma<<<(N + 15) / 16, 512, 0, stream>>>(x0, x0_out, W, bias, x0_out, N);
}